// TTAWrapper_34840774705472
// MI455X (gfx1250) — compile-verified
//
#include <hip/hip_runtime.h>
#include <cstdint>

// ---------------- problem constants (match reference) ----------------
#define N_DET 600
#define NP    608            // rows padded to 38*16 for WMMA M tiles
#define MT    38             // M tiles of 16
#define KSTEPS 150           // K=600 / 4 (wmma_f32_16x16x4)
#define KWORDS 5             // ceil(150/32) bitmap words per M tile
#define HW    (384 * 384)    // 147456 mask pixels, divisible by 128
#define KS    601            // LDS row stride (601 mod 64 = 25 -> bank-conflict-free)
#define SIGMA_F 0.5f
#define SCORE_THRESH_F 0.001f
#define MASK_IOU_THRESH_F 0.5f

typedef float v2f __attribute__((ext_vector_type(2)));
typedef float v8f __attribute__((ext_vector_type(8)));

#if defined(__has_builtin)
#if __has_builtin(__builtin_amdgcn_global_load_async_to_lds_b32) && \
    __has_builtin(__builtin_amdgcn_s_wait_asynccnt)
#define HAVE_ASYNC_LDS 1
#endif
#endif

// =====================================================================
// Kernel 1: sequential gaussian soft-NMS, one workgroup (20 wave32s).
// Writes keep (as 0/1 float) and decayed scores in ORIGINAL order.
// =====================================================================
__global__ __launch_bounds__(640) void softnms_kernel(
    const float* __restrict__ boxes, const float* __restrict__ scores,
    const int* __restrict__ labels, float* __restrict__ out_keep,
    float* __restrict__ out_dscore) {
  __shared__ float sb[N_DET][4];
  __shared__ float sarea[N_DET];
  __shared__ float ss[N_DET];
  __shared__ int   sorder[N_DET];
  __shared__ int   sactive[N_DET];
  __shared__ int   skeep[N_DET];
  __shared__ float sred[640];
  const int tid = threadIdx.x;

  // ---- max coordinate (for class shift) ----
  float m = -1e30f;
  for (int t = tid; t < N_DET * 4; t += 640) m = fmaxf(m, boxes[t]);
  sred[tid] = m;
  __syncthreads();
  if (tid == 0) {
    float mm = -1e30f;
    for (int t = 0; t < 640; ++t) mm = fmaxf(mm, sred[t]);
    sred[0] = mm + 1.0f;
  }
  __syncthreads();
  const float maxc = sred[0];

  // ---- stable rank for argsort(-scores) ----
  if (tid < N_DET) {
    const float si = scores[tid];
    int r = 0;
    for (int j = 0; j < N_DET; ++j) {
      const float sj = scores[j];
      if (sj > si || (sj == si && j < tid)) ++r;
    }
    sorder[r] = tid;
  }
  __syncthreads();

  // ---- gather sorted, class-shifted boxes ----
  if (tid < N_DET) {
    const int oi = sorder[tid];
    const float sh = (float)labels[oi] * maxc;
    const float x1 = boxes[oi * 4 + 0] + sh, y1 = boxes[oi * 4 + 1] + sh;
    const float x2 = boxes[oi * 4 + 2] + sh, y2 = boxes[oi * 4 + 3] + sh;
    sb[tid][0] = x1; sb[tid][1] = y1; sb[tid][2] = x2; sb[tid][3] = y2;
    sarea[tid] = (x2 - x1) * (y2 - y1);
    ss[tid] = scores[oi];
    sactive[tid] = 1;
    skeep[tid] = 0;
  }
  __syncthreads();

  // ---- sequential soft-NMS main loop ----
  for (int i = 0; i < N_DET; ++i) {
    const int act = sactive[i];
    const float bx1 = sb[i][0], by1 = sb[i][1], bx2 = sb[i][2], by2 = sb[i][3];
    const float ba = sarea[i];
    if (tid == i) skeep[i] = act;
    if (act && tid < N_DET && tid > i) {
      const float ix1 = fmaxf(bx1, sb[tid][0]), iy1 = fmaxf(by1, sb[tid][1]);
      const float ix2 = fminf(bx2, sb[tid][2]), iy2 = fminf(by2, sb[tid][3]);
      const float iw = fmaxf(ix2 - ix1, 0.f), ih = fmaxf(iy2 - iy1, 0.f);
      const float inter = iw * ih;
      const float iou = inter / (ba + sarea[tid] - inter);
      const float s2 = ss[tid] * expf(-(iou * iou) / SIGMA_F);  // decays even inactive j (matches ref)
      ss[tid] = s2;
      if (!(s2 > SCORE_THRESH_F)) sactive[tid] = 0;
    }
    __syncthreads();
  }

  // ---- scatter back to original order ----
  if (tid < N_DET) {
    const int oi = sorder[tid];
    out_keep[oi]   = skeep[tid] ? 1.0f : 0.0f;
    out_dscore[oi] = skeep[tid] ? ss[tid] : 0.0f;
  }
}

// =====================================================================
// Kernel 2: per-row same-class IoU>0.5 match count + use_avg flag.
// One block per detection row.
// =====================================================================
__global__ __launch_bounds__(256) void match_count_kernel(
    const float* __restrict__ boxes, const int* __restrict__ labels,
    const float* __restrict__ keep, int* __restrict__ cnt,
    int* __restrict__ use_avg) {
  const int i = blockIdx.x;
  __shared__ int sred[256];
  const int tid = threadIdx.x;
  const float ax1 = boxes[i * 4 + 0], ay1 = boxes[i * 4 + 1];
  const float ax2 = boxes[i * 4 + 2], ay2 = boxes[i * 4 + 3];
  const float aa = (ax2 - ax1) * (ay2 - ay1);
  const int li = labels[i];
  int c = 0;
  for (int j = tid; j < N_DET; j += 256) {
    if (labels[j] == li) {
      const float bx1 = boxes[j * 4 + 0], by1 = boxes[j * 4 + 1];
      const float bx2 = boxes[j * 4 + 2], by2 = boxes[j * 4 + 3];
      const float ba = (bx2 - bx1) * (by2 - by1);
      const float ix1 = fmaxf(ax1, bx1), iy1 = fmaxf(ay1, by1);
      const float ix2 = fminf(ax2, bx2), iy2 = fminf(ay2, by2);
      const float iw = fmaxf(ix2 - ix1, 0.f), ih = fmaxf(iy2 - iy1, 0.f);
      const float inter = iw * ih;
      if (inter / (aa + ba - inter) > MASK_IOU_THRESH_F) ++c;
    }
  }
  sred[tid] = c;
  __syncthreads();
  for (int s = 128; s > 0; s >>= 1) {
    if (tid < s) sred[tid] += sred[tid + s];
    __syncthreads();
  }
  if (tid == 0) {
    cnt[i] = sred[0];
    use_avg[i] = (keep[i] > 0.5f && sred[0] > 1) ? 1 : 0;
  }
}

// =====================================================================
// Kernel 3a/3b: build weight matrix Wmat [NP x 600] plus a 150-BIT
// nonzero bitmap per 16-row M-tile (bit ks set <=> K-step ks has any
// nonzero). out = Wmat @ flat realizes the entire select/average GEMM.
// =====================================================================
__global__ void kbits_init_kernel(unsigned* __restrict__ kbits) {
  const int t = blockIdx.x * blockDim.x + threadIdx.x;
  if (t < MT * KWORDS) kbits[t] = 0u;
}

__global__ __launch_bounds__(256) void build_w_kernel(
    const float* __restrict__ boxes, const int* __restrict__ labels,
    const int* __restrict__ cnt, const int* __restrict__ use_avg,
    float* __restrict__ Wmat, unsigned* __restrict__ kbits) {
  const int t = blockIdx.x * 256 + threadIdx.x;
  if (t >= NP * N_DET) return;
  const int i = t / N_DET, j = t - i * N_DET;
  float val = 0.f;
  if (i < N_DET) {
    if (use_avg[i]) {
      if (labels[i] == labels[j]) {
        const float ax1 = boxes[i * 4 + 0], ay1 = boxes[i * 4 + 1];
        const float ax2 = boxes[i * 4 + 2], ay2 = boxes[i * 4 + 3];
        const float bx1 = boxes[j * 4 + 0], by1 = boxes[j * 4 + 1];
        const float bx2 = boxes[j * 4 + 2], by2 = boxes[j * 4 + 3];
        const float aa = (ax2 - ax1) * (ay2 - ay1);
        const float ba = (bx2 - bx1) * (by2 - by1);
        const float ix1 = fmaxf(ax1, bx1), iy1 = fmaxf(ay1, by1);
        const float ix2 = fminf(ax2, bx2), iy2 = fminf(ay2, by2);
        const float iw = fmaxf(ix2 - ix1, 0.f), ih = fmaxf(iy2 - iy1, 0.f);
        const float inter = iw * ih;
        if (inter / (aa + ba - inter) > MASK_IOU_THRESH_F)
          val = 1.0f / (float)cnt[i];
      }
    } else {
      val = (i == j) ? 1.0f : 0.f;   // identity row -> out row = input mask row
    }
  }
  Wmat[t] = val;
  if (val != 0.f) {
    const int ks = j >> 2;
    atomicOr(&kbits[(i >> 4) * KWORDS + (ks >> 5)], 1u << (ks & 31));
  }
}

// =====================================================================
// Kernel 4: sparse-tile fp32 WMMA GEMM  out[600,HW] = Wmat[600,600] @ X.
// Block = 8 wave32s; each wave owns one 16x16 output tile.
// A block (16 x 600) staged once into LDS via CDNA5 async-to-LDS.
// Nonzero K-steps expanded into an LDS list once (thread 0, ctz scan);
// main loop is a counted for-loop, 2-deep software pipelined with
// DISJOINT register buffers (no rotation moves) so waits preceding each
// v_wmma cover only the elder in-flight loads.
// HBM floor: 708MB / 23.3TB/s ~30us.
// =====================================================================
__global__ __launch_bounds__(256) void masks_gemm_kernel(
    const float* __restrict__ Wmat, const float* __restrict__ X,
    const unsigned* __restrict__ kbits, float* __restrict__ out) {
  __shared__ float Alds[16 * KS];
  __shared__ int   ksList[KSTEPS];
  __shared__ int   snnz;
  const int nb   = blockIdx.x;            // 1152 column blocks of 128
  const int mt   = blockIdx.y;            // 38 row tiles of 16
  const int tid  = threadIdx.x;
  const int wave = tid >> 5;              // wave32
  const int lane = tid & 31;
  const int half = lane >> 4;             // 0: lanes 0-15, 1: lanes 16-31
  const int l16  = lane & 15;
  const int m0   = mt * 16;

  // ---- stage A tile (16 x 600) into LDS (async DMA path) ----
  const float* Wblk = Wmat + (size_t)m0 * N_DET;
  for (int idx = tid; idx < 16 * N_DET; idx += 256) {
    const int r = idx / N_DET, k = idx - r * N_DET;
#if HAVE_ASYNC_LDS
    __builtin_amdgcn_global_load_async_to_lds_b32(
        (__attribute__((address_space(1))) int*)(uintptr_t)(Wblk + idx),
        (__attribute__((address_space(3))) int*)(uint32_t)(uintptr_t)&Alds[r * KS + k],
        0, 0);
#else
    Alds[r * KS + k] = Wblk[idx];
#endif
  }

  // ---- expand K-step bitmap into an explicit list (thread 0) ----
  if (tid == 0) {
    const unsigned* kb = kbits + mt * KWORDS;
    int n = 0;
    for (int w = 0; w < KWORDS; ++w) {
      unsigned bits = kb[w];
      while (bits) {
        const int b = __builtin_ctz(bits);
        bits &= bits - 1u;
        ksList[n++] = w * 32 + b;
      }
    }
    snnz = n;
  }
#if HAVE_ASYNC_LDS
  __builtin_amdgcn_s_wait_asynccnt(0);
#endif
  __syncthreads();

  const int nnz = snnz;
  const int colBase = nb * 128 + wave * 16;
  const float* Xcol = X + colBase + l16;
  const float* Arow = &Alds[l16 * KS + 2 * half];

  // fragment loader: A from LDS (ds_load_2addr), B from global (2x b32)
  auto loadFrag = [&](int t, v2f& aF, v2f& bF) {
    const int ksv = ksList[t];
    const float* ap = Arow + ksv * 4;                      // K = 4ks+2h, +1
    aF.x = ap[0]; aF.y = ap[1];
    const float* bp = Xcol + (size_t)(ksv * 4 + 2 * half) * HW;
    bF.x = bp[0]; bF.y = bp[HW];
    __builtin_prefetch(bp + (size_t)2 * HW, 0, 1);         // global_prefetch_b8
  };

  v8f acc = {};
  v2f a0 = {}, b0 = {}, a1 = {}, b1 = {};
  if (nnz > 0) loadFrag(0, a0, b0);
  int t = 0;
  for (; t + 1 < nnz; t += 2) {
    loadFrag(t + 1, a1, b1);                               // fill buffer 1
    acc = __builtin_amdgcn_wmma_f32_16x16x4_f32(false, a0, false, b0,
                                                (short)0, acc, false, false);
    if (t + 2 < nnz) loadFrag(t + 2, a0, b0);              // refill buffer 0
    acc = __builtin_amdgcn_wmma_f32_16x16x4_f32(false, a1, false, b1,
                                                (short)0, acc, false, false);
  }
  if (t < nnz)
    acc = __builtin_amdgcn_wmma_f32_16x16x4_f32(false, a0, false, b0,
                                                (short)0, acc, false, false);

  // ---- store: VGPR r -> M = m0 + r + 8*half, N = colBase + l16 ----
  for (int r = 0; r < 8; ++r) {
    const int mrow = m0 + r + 8 * half;
    if (mrow < N_DET) out[(size_t)mrow * HW + colBase + l16] = acc[r];
  }
}

// =====================================================================
extern "C" void kernel_launch(void* const* d_in, const int* in_sizes, int n_in,
                              void* d_out, int out_size, void* d_ws,
                              size_t ws_size, hipStream_t stream) {
  (void)in_sizes; (void)n_in; (void)out_size; (void)ws_size;
  const float* boxes  = (const float*)d_in[0];
  const float* scores = (const float*)d_in[1];
  const float* masks  = (const float*)d_in[2];   // [600, HW]
  const int*   labels = (const int*)d_in[3];

  float* out        = (float*)d_out;
  float* out_keep   = out;                 // 600 (bool as 0/1 float)
  float* out_dscore = out + N_DET;         // 600
  float* out_masks  = out + 2 * N_DET;     // 600 * HW

  char* ws = (char*)d_ws;
  float*    Wmat   = (float*)ws;                            // NP*600 floats
  int*      cnt    = (int*)(ws + (size_t)NP * N_DET * sizeof(float));
  int*      useavg = cnt + N_DET;
  unsigned* kbits  = (unsigned*)(useavg + N_DET);           // MT*KWORDS words

  softnms_kernel<<<1, 640, 0, stream>>>(boxes, scores, labels, out_keep, out_dscore);
  match_count_kernel<<<N_DET, 256, 0, stream>>>(boxes, labels, out_keep, cnt, useavg);
  kbits_init_kernel<<<1, 256, 0, stream>>>(kbits);
  build_w_kernel<<<(NP * N_DET + 255) / 256, 256, 0, stream>>>(boxes, labels, cnt,
                                                               useavg, Wmat, kbits);
  masks_gemm_kernel<<<dim3(HW / 128, MT), 256, 0, stream>>>(Wmat, masks, kbits,
                                                            out_masks);
}